// LocalConv2DLayer_58660663328908
// MI455X (gfx1250) — compile-verified
//
#include <hip/hip_runtime.h>

// MI455X / gfx1250, wave32. FP32 WMMA (V_WMMA_F32_16X16X4_F32).
//
// res[b,oc] (60x60) = R (60x64 band) @ Ssum[b,oc] (64x64) @ R^T,
// where Ssum[y][x] = sum_c m^2(x[b,c,y,x], oc) is computed once per input
// pixel in VALU, and the all-ones 5x5 window reduction is done on the
// matrix pipes as two banded fp32 matmuls.
//
// Band fragments are generated in registers with a single unsigned range
// compare per element: R[row][k] = ((unsigned)(k - row) <= 4). The row/col
// validity (< 60) is intentionally omitted: padding rows/cols of T and Out
// only feed output elements that the bounds-checked store discards.

typedef __attribute__((ext_vector_type(2))) float v2f;
typedef __attribute__((ext_vector_type(8))) float v8f;

#define IN_C   3
#define OUT_C  32
#define HW     64
#define NOUT   60
#define PITCH  68   // LDS row pitch in floats (bank-conflict avoidance)

// Band matrix R[row][k] = 1 if row <= k <= row+4  (one unsigned compare)
__device__ __forceinline__ float band(int row, int k) {
    return ((unsigned)(k - row) <= 4u) ? 1.0f : 0.0f;
}

__global__ __launch_bounds__(256) void
localconv2d_wmma_kernel(const float* __restrict__ x,
                        const float* __restrict__ lb,
                        const float* __restrict__ rb,
                        float* __restrict__ out) {
    const int b   = blockIdx.x >> 5;   // batch
    const int oc  = blockIdx.x & 31;   // output channel

    __shared__ float S[HW * PITCH];    // Ssum (64x64, padded)
    __shared__ float T[HW * PITCH];    // T = R @ S  (64x64, padded)

    const int tid  = threadIdx.x;
    const int lane = tid & 31;
    const int wave = tid >> 5;

    // Bounds are constant across (c,ky,kx) for a given oc (broadcast linspace).
    const float l    = lb[oc * IN_C * 25];
    const float r    = rb[oc * IN_C * 25];
    const float dlr  = r - l;
    const float norm = 4.0f / (dlr * dlr);

    const float* xb = x + (size_t)b * IN_C * HW * HW;

    // ---- Phase 1: Ssum[y][x] = sum_c (clamp(p-l,0,1)*clamp(r-p,0,1)*norm)^2
    for (int idx = tid; idx < HW * HW; idx += 256) {
        float s = 0.0f;
        #pragma unroll
        for (int c = 0; c < IN_C; ++c) {
            float p  = xb[idx + c * HW * HW];
            float a  = fminf(fmaxf(p - l, 0.0f), 1.0f);  // relu(clip(p-l,-1,1))
            float bb = fminf(fmaxf(r - p, 0.0f), 1.0f);  // relu(clip(r-p,-1,1))
            float m  = a * bb * norm;
            s += m * m;
        }
        S[(idx >> 6) * PITCH + (idx & 63)] = s;
    }
    __syncthreads();

    // Tile assignment: 16 (4x4) output tiles, 8 waves, 2 adjacent tiles/wave
    // (the two tiles share the A fragment: same ti; tj and tj+1).
    const int ti    = wave >> 1;
    const int tj0   = (wave & 1) * 2;
    const int lrow  = lane & 15;
    const int khalf = (lane >> 4) << 1;       // lanes 0-15 -> K{0,1}; 16-31 -> K{2,3}
    const int rowA  = ti * 16 + lrow;
    const int rofs  = (lane >> 4) * 8;        // C/D: lanes 16-31 hold rows M+8

    // ---- Phase 2: T = R @ S  (A = band, generated in registers; B from LDS)
    {
        v8f d0 = {}; v8f d1 = {};
        #pragma unroll
        for (int kk = 0; kk < 16; ++kk) {
            const int k0 = kk * 4 + khalf;
            v2f a;
            a.x = band(rowA, k0);
            a.y = band(rowA, k0 + 1);
            const int col0 = tj0 * 16 + lrow;
            v2f b0, b1;
            b0.x = S[ k0      * PITCH + col0];
            b0.y = S[(k0 + 1) * PITCH + col0];
            b1.x = S[ k0      * PITCH + col0 + 16];
            b1.y = S[(k0 + 1) * PITCH + col0 + 16];
            d0 = __builtin_amdgcn_wmma_f32_16x16x4_f32(false, a, false, b0,
                                                       (short)0, d0, false, false);
            d1 = __builtin_amdgcn_wmma_f32_16x16x4_f32(false, a, false, b1,
                                                       (short)0, d1, false, false);
        }
        #pragma unroll
        for (int g = 0; g < 8; ++g) {
            const int rr = ti * 16 + g + rofs;
            T[rr * PITCH + tj0 * 16 + lrow]      = d0[g];
            T[rr * PITCH + tj0 * 16 + 16 + lrow] = d1[g];
        }
    }
    __syncthreads();

    // ---- Phase 3: Out = T @ R^T  (A from LDS; B = band^T in registers)
    {
        v8f e0 = {}; v8f e1 = {};
        #pragma unroll
        for (int kk = 0; kk < 16; ++kk) {
            const int k0 = kk * 4 + khalf;
            v2f a;
            a.x = T[rowA * PITCH + k0];
            a.y = T[rowA * PITCH + k0 + 1];
            const int j0 = tj0 * 16 + lrow;
            v2f b0, b1;
            b0.x = band(j0,      k0);          // R^T[k][j] = R[j][k]
            b0.y = band(j0,      k0 + 1);
            b1.x = band(j0 + 16, k0);
            b1.y = band(j0 + 16, k0 + 1);
            e0 = __builtin_amdgcn_wmma_f32_16x16x4_f32(false, a, false, b0,
                                                       (short)0, e0, false, false);
            e1 = __builtin_amdgcn_wmma_f32_16x16x4_f32(false, a, false, b1,
                                                       (short)0, e1, false, false);
        }
        float* outb = out + (size_t)(b * OUT_C + oc) * NOUT * NOUT;
        #pragma unroll
        for (int g = 0; g < 8; ++g) {
            const int rr  = ti * 16 + g + rofs;
            const int cc0 = tj0 * 16 + lrow;
            if (rr < NOUT) {
                if (cc0 < NOUT)      outb[rr * NOUT + cc0]      = e0[g];
                if (cc0 + 16 < NOUT) outb[rr * NOUT + cc0 + 16] = e1[g];
            }
        }
    }
}

extern "C" void kernel_launch(void* const* d_in, const int* in_sizes, int n_in,
                              void* d_out, int out_size, void* d_ws, size_t ws_size,
                              hipStream_t stream) {
    (void)in_sizes; (void)n_in; (void)out_size; (void)d_ws; (void)ws_size;
    const float* x  = (const float*)d_in[0];
    const float* lb = (const float*)d_in[1];
    const float* rb = (const float*)d_in[2];
    float* out = (float*)d_out;
    // One workgroup (8 wave32) per (batch, out_channel) pair: 16*32 = 512 blocks.
    localconv2d_wmma_kernel<<<dim3(16 * OUT_C), dim3(256), 0, stream>>>(x, lb, rb, out);
}